// Sparsemm_89111981457423
// MI455X (gfx1250) — compile-verified
//
#include <hip/hip_runtime.h>

#define FEAT 128

// ---------------- CDNA5 async global->LDS staging (probe via __has_builtin) ----
#if defined(__has_builtin)
#if __has_builtin(__builtin_amdgcn_global_load_async_to_lds_b64) && \
    __has_builtin(__builtin_amdgcn_s_wait_asynccnt)
#define HAVE_ASYNC_LDS 1
#endif
#endif
#ifndef HAVE_ASYNC_LDS
#define HAVE_ASYNC_LDS 0
#endif

typedef int v2i __attribute__((__vector_size__(2 * sizeof(int))));
typedef __attribute__((address_space(1))) v2i* v2i_gptr;
typedef __attribute__((address_space(3))) v2i* v2i_lptr;

// ---------------------------------------------------------------------------
// Phase 4 (hot kernel, placed first so the disasm snippet shows it):
// CSR SpMM, one wave32 per row; lane covers 4 contiguous features (float4).
// Edge metadata staged 32-at-a-time into per-wave LDS via the CDNA5 async
// global->LDS path; b-row gathers pipelined 4 edges deep to overlap L2 latency.
// ---------------------------------------------------------------------------
__global__ void spmm_csr_kernel(const int* __restrict__ row_start,
                                const int2* __restrict__ packed,
                                const float* __restrict__ b,
                                float* __restrict__ out, int n) {
    __shared__ int2 stage[8][32];  // 8 waves per 256-thread block
    const int wave = threadIdx.x >> 5;
    const int lane = threadIdx.x & 31;
    const int row = blockIdx.x * 8 + wave;
    if (row >= n) return;

    const int s = row_start[row];
    const int e = row_start[row + 1];
    float4 acc = make_float4(0.f, 0.f, 0.f, 0.f);
    const float* bl = b + (size_t)lane * 4;

    for (int cs = s; cs < e; cs += 32) {
        int m = e - cs;
        if (m > 32) m = 32;
#if HAVE_ASYNC_LDS
        if (lane < m) {
            __builtin_amdgcn_global_load_async_to_lds_b64(
                (v2i_gptr)(packed + cs + lane),
                (v2i_lptr)(&stage[wave][lane]),
                0, 0);
        }
        __builtin_amdgcn_s_wait_asynccnt(0);
#else
        if (lane < m) stage[wave][lane] = packed[cs + lane];
        // same-wave LDS ops are in-order; reads below observe these writes
#endif
        int j = 0;
        // 4-deep software pipeline: independent gathers form a memory clause
        for (; j + 4 <= m; j += 4) {
            int2 p0 = stage[wave][j + 0];
            int2 p1 = stage[wave][j + 1];
            int2 p2 = stage[wave][j + 2];
            int2 p3 = stage[wave][j + 3];
            const float4 r0 = *reinterpret_cast<const float4*>(bl + (size_t)p0.x * FEAT);
            const float4 r1 = *reinterpret_cast<const float4*>(bl + (size_t)p1.x * FEAT);
            const float4 r2 = *reinterpret_cast<const float4*>(bl + (size_t)p2.x * FEAT);
            const float4 r3 = *reinterpret_cast<const float4*>(bl + (size_t)p3.x * FEAT);
            const float v0 = __int_as_float(p0.y);
            const float v1 = __int_as_float(p1.y);
            const float v2 = __int_as_float(p2.y);
            const float v3 = __int_as_float(p3.y);
            acc.x = fmaf(v0, r0.x, acc.x); acc.y = fmaf(v0, r0.y, acc.y);
            acc.z = fmaf(v0, r0.z, acc.z); acc.w = fmaf(v0, r0.w, acc.w);
            acc.x = fmaf(v1, r1.x, acc.x); acc.y = fmaf(v1, r1.y, acc.y);
            acc.z = fmaf(v1, r1.z, acc.z); acc.w = fmaf(v1, r1.w, acc.w);
            acc.x = fmaf(v2, r2.x, acc.x); acc.y = fmaf(v2, r2.y, acc.y);
            acc.z = fmaf(v2, r2.z, acc.z); acc.w = fmaf(v2, r2.w, acc.w);
            acc.x = fmaf(v3, r3.x, acc.x); acc.y = fmaf(v3, r3.y, acc.y);
            acc.z = fmaf(v3, r3.z, acc.z); acc.w = fmaf(v3, r3.w, acc.w);
        }
        for (; j < m; ++j) {
            int2 p = stage[wave][j];
            float val = __int_as_float(p.y);
            const float4 brow = *reinterpret_cast<const float4*>(bl + (size_t)p.x * FEAT);
            acc.x = fmaf(val, brow.x, acc.x);
            acc.y = fmaf(val, brow.y, acc.y);
            acc.z = fmaf(val, brow.z, acc.z);
            acc.w = fmaf(val, brow.w, acc.w);
        }
    }
    *reinterpret_cast<float4*>(out + (size_t)row * FEAT + (size_t)lane * 4) = acc;
}

// ---------------------------------------------------------------------------
// Phase 0: small utility kernels
// ---------------------------------------------------------------------------
__global__ void fill_i32_kernel(int* __restrict__ p, int n, int v) {
    int i = blockIdx.x * blockDim.x + threadIdx.x;
    if (i < n) p[i] = v;
}

__global__ void copy_i32_kernel(const int* __restrict__ src, int* __restrict__ dst, int n) {
    int i = blockIdx.x * blockDim.x + threadIdx.x;
    if (i < n) dst[i] = src[i];
}

__global__ void zero_f4_kernel(float4* __restrict__ p, int n4) {
    int i = blockIdx.x * blockDim.x + threadIdx.x;
    int stride = gridDim.x * blockDim.x;
    float4 z = make_float4(0.f, 0.f, 0.f, 0.f);
    for (; i < n4; i += stride) p[i] = z;
}

// ---------------------------------------------------------------------------
// Phase 1: histogram of row ids into row_start[r+1]
// ---------------------------------------------------------------------------
__global__ void hist_kernel(const int* __restrict__ rows, int nnz, int* __restrict__ counts) {
    int e = blockIdx.x * blockDim.x + threadIdx.x;
    if (e < nnz) atomicAdd(&counts[rows[e] + 1], 1);
}

// ---------------------------------------------------------------------------
// Phase 2: single-block chunked inclusive scan over data[0..n-1] (n ~ 50001)
// ---------------------------------------------------------------------------
__global__ void scan_kernel(int* __restrict__ data, int n) {
    __shared__ int sh[1024];
    const int T = 1024;
    int carry = 0;
    for (int base = 0; base < n; base += T) {
        int i = base + (int)threadIdx.x;
        int v = (i < n) ? data[i] : 0;
        sh[threadIdx.x] = v;
        __syncthreads();
        // Hillis-Steele inclusive scan in LDS
        for (int off = 1; off < T; off <<= 1) {
            int t = (threadIdx.x >= (unsigned)off) ? sh[threadIdx.x - off] : 0;
            __syncthreads();
            sh[threadIdx.x] += t;
            __syncthreads();
        }
        if (i < n) data[i] = sh[threadIdx.x] + carry;
        int total = sh[T - 1];   // valid: last barrier above completed the scan
        __syncthreads();         // protect sh before next chunk overwrites it
        carry += total;
    }
}

// ---------------------------------------------------------------------------
// Phase 3: scatter edges into row-sorted packed (col, val_bits) pairs
// ---------------------------------------------------------------------------
__global__ void scatter_kernel(const int* __restrict__ rows, const int* __restrict__ cols,
                               const float* __restrict__ vals, int nnz,
                               int* __restrict__ cursor, int2* __restrict__ packed) {
    int e = blockIdx.x * blockDim.x + threadIdx.x;
    if (e >= nnz) return;
    int r = rows[e];
    int pos = atomicAdd(&cursor[r], 1);
    packed[pos] = make_int2(cols[e], __float_as_int(vals[e]));
}

// ---------------------------------------------------------------------------
// Fallback (no workspace): wave-per-edge gather + 4 global f32 atomics/lane
// ---------------------------------------------------------------------------
__global__ void spmm_atomic_kernel(const int* __restrict__ rows, const int* __restrict__ cols,
                                   const float* __restrict__ vals,
                                   const float* __restrict__ b,
                                   float* __restrict__ out, int nnz) {
    int wid = blockIdx.x * (blockDim.x >> 5) + (threadIdx.x >> 5);
    if (wid >= nnz) return;
    int lane = threadIdx.x & 31;
    int r = rows[wid];
    int c = cols[wid];
    float v = vals[wid];
    const float4 brow =
        *reinterpret_cast<const float4*>(b + (size_t)c * FEAT + (size_t)lane * 4);
    float* o = out + (size_t)r * FEAT + (size_t)lane * 4;
    atomicAdd(o + 0, v * brow.x);
    atomicAdd(o + 1, v * brow.y);
    atomicAdd(o + 2, v * brow.z);
    atomicAdd(o + 3, v * brow.w);
}

// ---------------------------------------------------------------------------
// Host launcher
// ---------------------------------------------------------------------------
extern "C" void kernel_launch(void* const* d_in, const int* in_sizes, int n_in,
                              void* d_out, int out_size, void* d_ws, size_t ws_size,
                              hipStream_t stream) {
    const int*   indices = (const int*)d_in[0];   // [2, NNZ] int32
    const float* values  = (const float*)d_in[1]; // [NNZ]
    const float* b       = (const float*)d_in[2]; // [N, 128]
    const int nnz = in_sizes[1];
    const int n   = in_sizes[2] / FEAT;
    const int* rows = indices;
    const int* cols = indices + nnz;
    float* out = (float*)d_out;

    // workspace layout: row_start (n+1 ints) | cursor (n ints) | packed (nnz int2)
    size_t off = 0;
    const size_t rs_off = off;  off += (size_t)(n + 1) * 4; off = (off + 15) & ~(size_t)15;
    const size_t cur_off = off; off += (size_t)n * 4;       off = (off + 15) & ~(size_t)15;
    const size_t pk_off = off;  off += (size_t)nnz * 8;
    const size_t need = off;

    if (d_ws != nullptr && ws_size >= need) {
        int*  row_start = (int*)((char*)d_ws + rs_off);
        int*  cursor    = (int*)((char*)d_ws + cur_off);
        int2* packed    = (int2*)((char*)d_ws + pk_off);

        fill_i32_kernel<<<(n + 1 + 255) / 256, 256, 0, stream>>>(row_start, n + 1, 0);
        hist_kernel<<<(nnz + 255) / 256, 256, 0, stream>>>(rows, nnz, row_start);
        scan_kernel<<<1, 1024, 0, stream>>>(row_start, n + 1);
        copy_i32_kernel<<<(n + 255) / 256, 256, 0, stream>>>(row_start, cursor, n);
        scatter_kernel<<<(nnz + 255) / 256, 256, 0, stream>>>(rows, cols, values, nnz,
                                                              cursor, packed);
        spmm_csr_kernel<<<(n + 7) / 8, 256, 0, stream>>>(row_start, packed, b, out, n);
    } else {
        zero_f4_kernel<<<4096, 256, 0, stream>>>((float4*)out, out_size / 4);
        spmm_atomic_kernel<<<(nnz + 7) / 8, 256, 0, stream>>>(rows, cols, values, b, out, nnz);
    }
}